// NousModel_50620484551275
// MI455X (gfx1250) — compile-verified
//
#include <hip/hip_runtime.h>

#define NC 33          // coefficients per polynomial
#define NR 32          // roots per polynomial == wave32 lanes
#define WAVES 8        // polynomials (waves) per block
#define BLOCK (WAVES * 32)

typedef float        v2f __attribute__((ext_vector_type(2)));
typedef float        v8f __attribute__((ext_vector_type(8)));
typedef unsigned int v4u __attribute__((ext_vector_type(4)));
typedef int          v4i __attribute__((ext_vector_type(4)));
typedef int          v8i __attribute__((ext_vector_type(8)));

__device__ __forceinline__ float lane_bcast(float x, int srcLane) {
  return __builtin_bit_cast(
      float, __builtin_amdgcn_ds_bpermute(srcLane << 2, __builtin_bit_cast(int, x)));
}

__global__ __launch_bounds__(BLOCK) void dk_solve_kernel(
    const float* __restrict__ coeffs, float* __restrict__ out, int npoly) {
  __shared__ float sCoef[WAVES][36];      // TDM lands 33 floats per wave here
  __shared__ float sZr[WAVES][NR];
  __shared__ float sZi[WAVES][NR];
  __shared__ float sClu[WAVES][3][NR];    // [sumRe, sumIm, count] per root

  const int lane = threadIdx.x & 31;
  const int wave = threadIdx.x >> 5;
  int poly = blockIdx.x * WAVES + wave;
  if (poly >= npoly) poly = npoly - 1;    // duplicate last poly; identical writes are benign

  // ------- Stage coefficients into LDS via the Tensor Data Mover (TDM) -------
  // D# group0: count=1 | lds_addr | global_addr[56:0] | type=2
  // D# group1: data_size=4B, tensor_dim0=33, tensor_dim1=1, tile_dim0=33, stride0=33
  {
    const int polyU = __builtin_amdgcn_readfirstlane(poly);
    const unsigned long long ga =
        (unsigned long long)(const void*)(coeffs + (unsigned long long)polyU * NC);
    unsigned ldsOff = (unsigned)(unsigned long long)(void*)&sCoef[wave][0];
    ldsOff = (unsigned)__builtin_amdgcn_readfirstlane((int)ldsOff);

    v4u g0;
    g0.x = 1u;                                                 // count=1, user descriptor
    g0.y = ldsOff;                                             // lds_addr (bytes)
    g0.z = (unsigned)ga;                                       // global_addr[31:0]
    g0.w = ((unsigned)(ga >> 32) & 0x01FFFFFFu) | (2u << 30);  // addr[56:32] | type=2
    v8i g1;
    g1[0] = (2 << 16);                                         // data_size = 4 bytes
    g1[1] = (NC << 16);                                        // tensor_dim0 lo16 = 33
    g1[2] = (1 << 16);                                         // dim0 hi=0 | tensor_dim1 = 1
    g1[3] = (NC << 16);                                        // dim1 hi=0 | tile_dim0 = 33
    g1[4] = 0;                                                 // tile_dim1/2 unused
    g1[5] = NC;                                                // tensor_dim0_stride lo32
    g1[6] = 0;
    g1[7] = 0;
    // <=2-D tensor: VADDR2/VADDR3 are NULL -> 2-group short form.
    asm volatile("tensor_load_to_lds %0, %1" : : "s"(g0), "s"(g1) : "memory");
#if __has_builtin(__builtin_amdgcn_s_wait_tensorcnt)
    __builtin_amdgcn_s_wait_tensorcnt(0);
#else
    asm volatile("s_wait_tensorcnt 0x0" ::: "memory");
#endif
    __asm__ volatile("" ::: "memory");   // keep LDS reads below the tensor wait
  }

  // ---------------- Monic normalization, coefficients live in VGPRs ----------
  float c[NC];
  {
    const float inv0 = 1.0f / sCoef[wave][0];
#pragma unroll
    for (int i = 0; i < NC; ++i) c[i] = sCoef[wave][i] * inv0;
  }

  // ---------------- Initial guesses: perturbed circle ------------------------
  const float TWO_PI = 6.28318530717958647692f;
  float ang = (float)lane * (TWO_PI / (float)NR)
            + 0.01f * sinf((float)lane * 12.9898f + 4.1414f);
  float radius = 0.5f + 0.6f * powf(fabsf(c[1]), 1.0f / (float)NR);
  float zr = radius * cosf(ang);
  float zi = radius * sinf(ang);

  // ---------------- Phase 1: Durand-Kerner (60 iterations) -------------------
  for (int it = 0; it < 60; ++it) {
    float pr = 0.0f, pim = 0.0f;
#pragma unroll
    for (int i = 0; i < NC; ++i) {           // p = p*z + c[i]
      float t = pr * zr - pim * zi + c[i];
      pim = pr * zi + pim * zr;
      pr = t;
    }
    float dr = 1.0f, di = 0.0f;              // prod_{k!=j} (z_j - z_k)
    for (int s = 1; s < NR; ++s) {
      int src = (lane + s) & (NR - 1);
      float orr = lane_bcast(zr, src);
      float oii = lane_bcast(zi, src);
      float er = zr - orr, ei = zi - oii;
      float nr2 = dr * er - di * ei;
      di = dr * ei + di * er;
      dr = nr2;
    }
    dr += 1e-30f;
    float inv = 1.0f / (dr * dr + di * di);
    float cr = (pr * dr + pim * di) * inv;
    float ci = (pim * dr - pr * di) * inv;
    zr -= cr; zi -= ci;
  }

  // ---------------- Phase 2: multiplicity-aware Newton (30 iterations) -------
  for (int it = 0; it < 30; ++it) {
    float pr = 0.f, pim = 0.f, qr = 0.f, qi = 0.f, s2r = 0.f, s2i = 0.f;
#pragma unroll
    for (int i = 0; i < NC; ++i) {
      float t;
      t = s2r * zr - s2i * zi + 2.0f * qr;  s2i = s2r * zi + s2i * zr + 2.0f * qi;  s2r = t;
      t = qr * zr - qi * zi + pr;           qi  = qr * zi + qi * zr + pim;          qr  = t;
      t = pr * zr - pim * zi + c[i];        pim = pr * zi + pim * zr;               pr  = t;
    }
    float numr = pr * s2r - pim * s2i;
    float numi = pr * s2i + pim * s2r;
    float denr = (qr * qr - qi * qi) - numr;
    float deni = (2.0f * qr * qi) - numi;
    float dmag2 = denr * denr + deni * deni;
    float m = (sqrtf(dmag2) < 1e-30f)
                  ? 1.0f
                  : fminf(fmaxf((numr * denr + numi * deni) / dmag2 + 1.0f, 1.0f), (float)NR);
    float sdr = qr, sdi = qi;
    if (sqrtf(qr * qr + qi * qi) < 1e-30f) { sdr = 1e-30f; sdi = 0.0f; }
    float inv = 1.0f / (sdr * sdr + sdi * sdi);
    float cr = m * (pr * sdr + pim * sdi) * inv;
    float ci = m * (pim * sdr - pr * sdi) * inv;
    zr -= cr; zi -= ci;
  }

  // ---------------- Clustering: einsum('ij,j->i', near, z) via WMMA ----------
  const float THR2 = 1e-12f;                 // (1e-6)^2
  sZr[wave][lane] = zr;
  sZi[wave][lane] = zi;
  __syncthreads();
#if __has_builtin(__builtin_amdgcn_wmma_f32_16x16x4_f32)
  {
    const int row0 = lane & 15;
    const int khalf = (lane >> 4) & 1;       // lanes 16-31 supply K+2, K+3
    const int ncol = lane & 15;
#pragma unroll
    for (int t = 0; t < 2; ++t) {            // M tiles: rows 0-15, 16-31
      v8f acc = {0, 0, 0, 0, 0, 0, 0, 0};
      const int row = t * 16 + row0;
      const float rr = sZr[wave][row], ri = sZi[wave][row];
#pragma unroll
      for (int kc = 0; kc < 8; ++kc) {       // K chunks of 4
        const int k0 = kc * 4 + khalf * 2;
        float z0r = sZr[wave][k0],     z0i = sZi[wave][k0];
        float z1r = sZr[wave][k0 + 1], z1i = sZi[wave][k0 + 1];
        float d0r = rr - z0r, d0i = ri - z0i;
        float d1r = rr - z1r, d1i = ri - z1i;
        // A = near matrix (0/1); B columns: [z.re, z.im, 1, 0...]
        v2f A, B;
        A.x = (d0r * d0r + d0i * d0i < THR2) ? 1.0f : 0.0f;
        A.y = (d1r * d1r + d1i * d1i < THR2) ? 1.0f : 0.0f;
        B.x = (ncol == 0) ? z0r : (ncol == 1) ? z0i : (ncol == 2) ? 1.0f : 0.0f;
        B.y = (ncol == 0) ? z1r : (ncol == 1) ? z1i : (ncol == 2) ? 1.0f : 0.0f;
        acc = __builtin_amdgcn_wmma_f32_16x16x4_f32(false, A, false, B,
                                                    (short)0, acc, false, false);
      }
      // C layout: VGPR r -> M = r (+8 for lanes>=16), N = lane&15
      const int col = lane & 15;
      const int mb = t * 16 + ((lane & 16) ? 8 : 0);
      if (col < 3) {
#pragma unroll
        for (int r = 0; r < 8; ++r) sClu[wave][col][mb + r] = acc[r];
      }
    }
  }
#else
  {
    float sumr = 0.f, sumi = 0.f, cnt = 0.f;
    for (int k = 0; k < NR; ++k) {
      float okr = sZr[wave][k], oki = sZi[wave][k];
      float er = zr - okr, ei = zi - oki;
      float nearf = (er * er + ei * ei < THR2) ? 1.0f : 0.0f;
      sumr += nearf * okr; sumi += nearf * oki; cnt += nearf;
    }
    sClu[wave][0][lane] = sumr; sClu[wave][1][lane] = sumi; sClu[wave][2][lane] = cnt;
  }
#endif
  __syncthreads();
  {
    float sumr = sClu[wave][0][lane];
    float sumi = sClu[wave][1][lane];
    float cntf = sClu[wave][2][lane];
    int cnt = (int)(cntf + 0.5f);
    if (cnt > 1) { zr = sumr / cntf; zi = sumi / cntf; }
    if (cnt == NR) { zr = -c[1] / (float)NR; zi = 0.0f; }
  }

  // ---------------- Phase 3: polishing with pair multiplicity (20 iterations)
  for (int it = 0; it < 20; ++it) {
    float pr = 0.f, pim = 0.f, qr = 0.f, qi = 0.f;
#pragma unroll
    for (int i = 0; i < NC; ++i) {
      float t;
      t = qr * zr - qi * zi + pr;    qi  = qr * zi + qi * zr + pim;  qr = t;
      t = pr * zr - pim * zi + c[i]; pim = pr * zi + pim * zr;       pr = t;
    }
    float md2 = 3.4e38f;
    for (int s = 1; s < NR; ++s) {
      int src = (lane + s) & (NR - 1);
      float orr = lane_bcast(zr, src);
      float oii = lane_bcast(zi, src);
      float er = zr - orr, ei = zi - oii;
      md2 = fminf(md2, er * er + ei * ei);
    }
    float mult = (md2 < THR2) ? 2.0f : 1.0f;
    float sdr = qr, sdi = qi;
    if (sqrtf(qr * qr + qi * qi) < 1e-30f) { sdr = 1e-30f; sdi = 0.0f; }
    float inv = 1.0f / (sdr * sdr + sdi * sdi);
    float cr = mult * (pr * sdr + pim * sdi) * inv;
    float ci = mult * (pim * sdr - pr * sdi) * inv;
    zr -= cr; zi -= ci;
  }

  // ---------------- Output: [b, n, 2] view_as_real, single b64 store ---------
  float2 res;
  res.x = zr;
  res.y = zi;
  *(float2*)(out + (unsigned long long)poly * (NR * 2) + lane * 2) = res;
}

extern "C" void kernel_launch(void* const* d_in, const int* in_sizes, int n_in,
                              void* d_out, int out_size, void* d_ws, size_t ws_size,
                              hipStream_t stream) {
  const float* coeffs = (const float*)d_in[0];
  float* out = (float*)d_out;
  const int npoly = in_sizes[0] / NC;                 // 16384
  const int blocks = (npoly + WAVES - 1) / WAVES;     // 2048 blocks x 256 threads
  dk_solve_kernel<<<blocks, BLOCK, 0, stream>>>(coeffs, out, npoly);
}